// QuadrupletLoss_16621523436336
// MI455X (gfx1250) — compile-verified
//
#include <hip/hip_runtime.h>
#include <math.h>

// QuadrupletLoss NT-Xent, fused:
//   Kernel 1: d = (A @ T^T) * 20 tile-wise via V_WMMA_F32_16X16X4_F32 with the
//             masked online-softmax row stats fused in the epilogue. Two
//             independent K-chains (16 WMMAs each) for intra-wave matrix-pipe
//             ILP; columns split 8-ways across blocks for occupancy (4096
//             waves). Emits partial (rowmax, expsum) pairs.
//   Kernel 2: merge the 8 column-partials per row, then per-class sums/counts
//             (labels < 50) in LDS -> closed-form scalar loss.

typedef __attribute__((ext_vector_type(2))) float v2f;
typedef __attribute__((ext_vector_type(8))) float v8f;

#define DIMK    128
#define KSTEPS  (DIMK / 4)     // 32 WMMA K-steps total
#define KHALF   (KSTEPS / 2)   // 16 per chain
#define CSPLIT  8              // column-range split (occupancy)
#define TEMP_INV 20.0f         // 1 / 0.05 (reference hardcodes temp = 0.05)

__global__ void __launch_bounds__(32)
rowstats_kernel(const float* __restrict__ A, const float* __restrict__ T,
                const int* __restrict__ labels,
                float* __restrict__ m_part, float* __restrict__ p_part, int B)
{
    const int lane    = threadIdx.x;   // wave32
    const int l15     = lane & 15;
    const int half    = lane >> 4;     // 0: K+{0,1}, 1: K+{2,3} per 4-wide K step
    const int rowbase = blockIdx.x * 16;
    const int colspan = B / CSPLIT;
    const int colbase = blockIdx.y * colspan;

    // A fragment (16x4 f32 layout): lane holds row rowbase+l15,
    // VGPR pair = K = ks*4 + 2*half + {0,1}. Cache the full K=128 slab in regs.
    v2f afrag[KSTEPS];
    const float* arow = A + (size_t)(rowbase + l15) * DIMK + 2 * half;
#pragma unroll
    for (int ks = 0; ks < KSTEPS; ++ks)
        afrag[ks] = *(const v2f*)(arow + ks * 4);

    // Labels of the 8 rows this lane's accumulator VGPRs correspond to
    // (C/D layout: VGPR r -> M = r + 8*half).
    int rlab[8];
#pragma unroll
    for (int r = 0; r < 8; ++r)
        rlab[r] = labels[rowbase + 8 * half + r];

    // Per-lane online softmax state over the columns this lane sees.
    float m[8], p[8];
#pragma unroll
    for (int r = 0; r < 8; ++r) { m[r] = -1e30f; p[r] = 0.0f; }

    const int ntiles = colspan >> 4;
    for (int ct = 0; ct < ntiles; ++ct) {
        const int   col  = colbase + ct * 16 + l15;
        const int   lcol = labels[col];
        // B fragment (4x16 f32 layout): B[k][n] = T[n][k]; lane holds column
        // `col`, K = ks*4 + 2*half + {0,1} -- same addressing pattern as A.
        const float* trow = T + (size_t)col * DIMK + 2 * half;

        // Two independent accumulator chains -> matrix-pipe ILP.
        v8f acc0 = {}, acc1 = {};
#pragma unroll
        for (int ks = 0; ks < KHALF; ++ks) {
            v2f b0 = *(const v2f*)(trow + ks * 4);
            v2f b1 = *(const v2f*)(trow + (ks + KHALF) * 4);
            acc0 = __builtin_amdgcn_wmma_f32_16x16x4_f32(
                false, afrag[ks], false, b0, (short)0, acc0, false, false);
            acc1 = __builtin_amdgcn_wmma_f32_16x16x4_f32(
                false, afrag[ks + KHALF], false, b1, (short)0, acc1, false, false);
        }

        // Fused epilogue: masked online max / expsum per row.
#pragma unroll
        for (int r = 0; r < 8; ++r) {
            if (lcol != rlab[r]) {
                float v = (acc0[r] + acc1[r]) * TEMP_INV;
                if (v > m[r]) {
                    p[r] = p[r] * __expf(m[r] - v) + 1.0f;
                    m[r] = v;
                } else {
                    p[r] += __expf(v - m[r]);
                }
            }
        }
    }

    // Merge (max, expsum) pairs across the 16 lanes of each half-wave.
#pragma unroll
    for (int r = 0; r < 8; ++r) {
#pragma unroll
        for (int off = 1; off < 16; off <<= 1) {
            float m2 = __shfl_xor(m[r], off, 16);
            float p2 = __shfl_xor(p[r], off, 16);
            float mm = fmaxf(m[r], m2);
            p[r] = p[r] * __expf(m[r] - mm) + p2 * __expf(m2 - mm);
            m[r] = mm;
        }
    }

    if (l15 == 0) {
        const size_t base = (size_t)blockIdx.y * B + rowbase + 8 * half;
#pragma unroll
        for (int r = 0; r < 8; ++r) {
            m_part[base + r] = m[r];
            p_part[base + r] = p[r];
        }
    }
}

__global__ void __launch_bounds__(1024)
finalize_kernel(const float* __restrict__ m_part, const float* __restrict__ p_part,
                const int* __restrict__ labels, float* __restrict__ out, int B)
{
    __shared__ float clsSum[64];
    __shared__ int   clsCnt[64];
    const int t = threadIdx.x;
    if (t < 64) { clsSum[t] = 0.0f; clsCnt[t] = 0; }
    __syncthreads();

    for (int i = t; i < B; i += (int)blockDim.x) {
        // Merge the CSPLIT column-partials for row i.
        float mv[CSPLIT], pv[CSPLIT];
        float mm = -1e30f;
#pragma unroll
        for (int k = 0; k < CSPLIT; ++k) {
            mv[k] = m_part[(size_t)k * B + i];
            pv[k] = p_part[(size_t)k * B + i];
            mm = fmaxf(mm, mv[k]);
        }
        float s = 0.0f;
#pragma unroll
        for (int k = 0; k < CSPLIT; ++k)
            s += pv[k] * __expf(mv[k] - mm);

        const int c = labels[i];
        atomicAdd(&clsSum[c], s);   // ds_add_f32
        atomicAdd(&clsCnt[c], 1);   // ds_add_u32
    }
    __syncthreads();

    if (t == 0) {
        float tot = 0.0f;
        for (int c = 0; c < 64; ++c) tot += clsSum[c];
        float acc = 0.0f;
        for (int c = 0; c < 64; ++c) {
            const int cnt = clsCnt[c];
            if (cnt == 0) continue;
            const float nds  = tot - clsSum[c];      // sum of s over diff-label rows
            const int   negc = B - cnt;              // negcount for this class
            const float x    = (negc > 0) ? (nds / (float)negc) : nds;
            if (cnt > 1)                              // mask_sum = cnt-1 > 0
                acc += (float)cnt * logf(x + 1e-12f);
        }
        // loss = -mean(pos_logits) = (1/B) * sum_c cnt*[cnt>1]*log(x_c + 1e-12)
        out[0] = acc / (float)B;
    }
}

extern "C" void kernel_launch(void* const* d_in, const int* in_sizes, int n_in,
                              void* d_out, int out_size, void* d_ws, size_t ws_size,
                              hipStream_t stream) {
    (void)n_in; (void)out_size; (void)ws_size;
    const float* anchor = (const float*)d_in[0];
    const float* target = (const float*)d_in[1];
    const int*   labels = (const int*)d_in[2];
    const int B = in_sizes[2];          // 8192; D fixed at 128

    float* m_part = (float*)d_ws;                       // [CSPLIT * B]
    float* p_part = m_part + (size_t)CSPLIT * B;        // [CSPLIT * B]

    dim3 grid(B / 16, CSPLIT);
    rowstats_kernel<<<grid, 32, 0, stream>>>(anchor, target, labels,
                                             m_part, p_part, B);
    finalize_kernel<<<1, 1024, 0, stream>>>(m_part, p_part, labels,
                                            (float*)d_out, B);
}